// GCNConvNet_51754355916835
// MI455X (gfx1250) — compile-verified
//
#include <hip/hip_runtime.h>
#include <math.h>

// ---------------- problem constants (match reference) ----------------
#define N_NODES  50000
#define N_EDGES  500000
#define F_IN     128
#define HID      160
#define F_OUT    128
#define N_GRAPHS 50
#define OUT_DIM  10

#define KMAX   160          // max K over all layers
#define LDS_S  (KMAX + 4)   // padded LDS row stride (floats) to spread banks

typedef __attribute__((ext_vector_type(2))) float v2f;
typedef __attribute__((ext_vector_type(8))) float v8f;

// ---------------- small utility kernels ----------------
__global__ void fill_f32(float* __restrict__ p, float v, int n) {
    int i = blockIdx.x * blockDim.x + threadIdx.x;
    if (i < n) p[i] = v;
}

__global__ void deg_kernel(const int* __restrict__ dst, float* __restrict__ deg, int E) {
    int e = blockIdx.x * blockDim.x + threadIdx.x;
    if (e < E) atomicAdd(&deg[dst[e]], 1.0f);
}

__global__ void dinv_kernel(float* __restrict__ deg, int n) {
    int i = blockIdx.x * blockDim.x + threadIdx.x;
    if (i < n) deg[i] = rsqrtf(deg[i] + 2.0f);   // improved GCN: +2 self-loop weight
}

// ---------------- WMMA fp32 GEMM: XW = H[N,K] @ W[K,F] ----------------
// Block = 256 threads = 8 wave32s; one block owns a 16-row stripe of H.
// The 16xK A stripe (<=10 KB) is staged once into LDS and shared by all
// waves/column-tiles (CDNA5 WGP has 320 KB LDS; this removes ~10x redundant
// global A traffic). Each wave computes 16x16 output tiles with
// V_WMMA_F32_16X16X4_F32 (exact fp32, matches the fp32 reference).
//
// A (16x4, 32-bit): lanes 0-15 -> M=lane, regs {K=0,K=1}; lanes 16-31 -> {K=2,K=3}.
// B (4x16, 32-bit): mirrored: lanes 0-15 -> N=lane, regs {K=0,K=1}; lanes 16-31 -> {K=2,K=3}.
// C/D (16x16 f32): reg r -> M = r + (lane>=16 ? 8 : 0), N = lane&15.
__global__ __launch_bounds__(256)
void gcn_gemm_wmma(const float* __restrict__ H, const float* __restrict__ W,
                   float* __restrict__ XW, int K, int F) {
    __shared__ float As[16 * LDS_S];

    const int tid  = threadIdx.x;
    const int lane = tid & 31;
    const int wave = tid >> 5;                  // 8 waves / block (wave32)
    const int row0 = blockIdx.x << 4;           // N_NODES % 16 == 0
    const int m    = lane & 15;
    const int half = lane >> 4;                 // 0: K=0,1   1: K=2,3
    const int kq   = half * 2;

    // cooperative stage of the 16 x K A stripe into LDS
    for (int i = tid; i < 16 * K; i += 256) {
        int r = i / K;
        int c = i - r * K;
        As[r * LDS_S + c] = H[(size_t)(row0 + r) * K + c];
    }
    __syncthreads();

    const float* __restrict__ arow = As + m * LDS_S;
    const int nColTiles = F >> 4;

    for (int ct = wave; ct < nColTiles; ct += 8) {
        const int col0 = ct << 4;
        const float* __restrict__ wcol = W + col0 + m;   // n == lane&15
        v8f acc = {};
        for (int k0 = 0; k0 < K; k0 += 4) {
            v2f a, b;
            a.x = arow[k0 + kq];                         // ds_load from LDS
            a.y = arow[k0 + kq + 1];
            b.x = wcol[(size_t)(k0 + kq)     * F];       // W resident in WGP$/L2
            b.y = wcol[(size_t)(k0 + kq + 1) * F];
            acc = __builtin_amdgcn_wmma_f32_16x16x4_f32(
                      /*neg_a=*/false, a, /*neg_b=*/false, b,
                      /*c_mod=*/(short)0, acc, /*reuse_a=*/false, /*reuse_b=*/false);
        }
#pragma unroll
        for (int r = 0; r < 8; ++r) {
            int mo = r + half * 8;
            XW[(size_t)(row0 + mo) * F + col0 + m] = acc[r];
        }
    }
}

// agg = 2*dinv^2 * xw + bias   (full overwrite -> also clears workspace poison)
__global__ void selfloop_bias_kernel(const float* __restrict__ xw,
                                     const float* __restrict__ dinv,
                                     const float* __restrict__ bias,
                                     float* __restrict__ agg, int F) {
    const int fq = F >> 2;
    int idx = blockIdx.x * blockDim.x + threadIdx.x;
    if (idx >= N_NODES * fq) return;
    int n = idx / fq;
    int f = (idx - n * fq) << 2;
    float di = dinv[n];
    float c  = 2.0f * di * di;
    float4 v = *(const float4*)(xw + (size_t)n * F + f);
    float4 b = *(const float4*)(bias + f);
    float4 o = make_float4(c * v.x + b.x, c * v.y + b.y, c * v.z + b.z, c * v.w + b.w);
    *(float4*)(agg + (size_t)n * F + f) = o;
}

// per-(edge, feature4) scatter-add of normalized messages; L2-resident atomics
__global__ void scatter_kernel(const float* __restrict__ xw,
                               const int* __restrict__ src,
                               const int* __restrict__ dst,
                               const float* __restrict__ dinv,
                               float* __restrict__ agg, int F) {
    const int fq = F >> 2;
    int idx = blockIdx.x * blockDim.x + threadIdx.x;
    if (idx >= N_EDGES * fq) return;
    int e = idx / fq;
    int f = (idx - e * fq) << 2;
    int s = src[e], d = dst[e];
    float nrm = dinv[s] * dinv[d];
    float4 v = *(const float4*)(xw + (size_t)s * F + f);
    float* o = agg + (size_t)d * F + f;
    atomicAdd(o + 0, v.x * nrm);
    atomicAdd(o + 1, v.y * nrm);
    atomicAdd(o + 2, v.z * nrm);
    atomicAdd(o + 3, v.w * nrm);
}

// ---------------- pooling + head ----------------
__global__ void count_kernel(const int* __restrict__ batch, float* __restrict__ cnt) {
    int n = blockIdx.x * blockDim.x + threadIdx.x;
    if (n < N_NODES) atomicAdd(&cnt[batch[n]], 1.0f);
}

__global__ void pool_kernel(const float* __restrict__ h, const int* __restrict__ batch,
                            float* __restrict__ pool) {
    const int fq = F_OUT >> 2;
    int idx = blockIdx.x * blockDim.x + threadIdx.x;
    if (idx >= N_NODES * fq) return;
    int n = idx / fq;
    int f = (idx - n * fq) << 2;
    int g = batch[n];
    float4 v = *(const float4*)(h + (size_t)n * F_OUT + f);
    float* o = pool + (size_t)g * F_OUT + f;
    atomicAdd(o + 0, v.x);
    atomicAdd(o + 1, v.y);
    atomicAdd(o + 2, v.z);
    atomicAdd(o + 3, v.w);
}

__global__ void head_kernel(const float* __restrict__ pool, const float* __restrict__ cnt,
                            const float* __restrict__ fcw, const float* __restrict__ fcb,
                            float* __restrict__ logits) {
    int t = blockIdx.x * blockDim.x + threadIdx.x;
    if (t >= N_GRAPHS * OUT_DIM) return;
    int g = t / OUT_DIM, o = t - g * OUT_DIM;
    float inv = 1.0f / fmaxf(cnt[g], 1.0f);
    float acc = fcb[o];
    for (int f = 0; f < F_OUT; ++f)
        acc = fmaf(pool[g * F_OUT + f] * inv, fcw[f * OUT_DIM + o], acc);
    logits[t] = acc;
}

__global__ void logsoftmax_kernel(const float* __restrict__ logits, float* __restrict__ out) {
    int g = blockIdx.x * blockDim.x + threadIdx.x;
    if (g >= N_GRAPHS) return;
    float mx = -3.4e38f;
    for (int o = 0; o < OUT_DIM; ++o) mx = fmaxf(mx, logits[g * OUT_DIM + o]);
    float s = 0.0f;
    for (int o = 0; o < OUT_DIM; ++o) s += expf(logits[g * OUT_DIM + o] - mx);
    float lse = mx + logf(s);
    for (int o = 0; o < OUT_DIM; ++o) out[g * OUT_DIM + o] = logits[g * OUT_DIM + o] - lse;
}

// ---------------- host-side orchestration ----------------
static inline int cdiv(int a, int b) { return (a + b - 1) / b; }

static void run_layer(hipStream_t stream, const float* Hin, const float* Wmat,
                      const float* bias, int K, int F, float* XW, float* AGG,
                      const int* src, const int* dst, const float* dinv) {
    gcn_gemm_wmma<<<N_NODES / 16, 256, 0, stream>>>(Hin, Wmat, XW, K, F);
    int initN = N_NODES * (F >> 2);
    selfloop_bias_kernel<<<cdiv(initN, 256), 256, 0, stream>>>(XW, dinv, bias, AGG, F);
    int scatN = N_EDGES * (F >> 2);
    scatter_kernel<<<cdiv(scatN, 256), 256, 0, stream>>>(XW, src, dst, dinv, AGG, F);
}

extern "C" void kernel_launch(void* const* d_in, const int* in_sizes, int n_in,
                              void* d_out, int out_size, void* d_ws, size_t ws_size,
                              hipStream_t stream) {
    (void)in_sizes; (void)n_in; (void)out_size; (void)ws_size;

    const float* x     = (const float*)d_in[0];
    const int*   ei    = (const int*)d_in[1];         // [2, E] row-major
    const int*   src   = ei;
    const int*   dst   = ei + N_EDGES;
    const int*   batch = (const int*)d_in[2];
    const float* W0 = (const float*)d_in[3];  const float* b0 = (const float*)d_in[4];
    const float* W1 = (const float*)d_in[5];  const float* b1 = (const float*)d_in[6];
    const float* W2 = (const float*)d_in[7];  const float* b2 = (const float*)d_in[8];
    const float* W3 = (const float*)d_in[9];  const float* b3 = (const float*)d_in[10];
    const float* fcw = (const float*)d_in[11]; const float* fcb = (const float*)d_in[12];
    float* out = (float*)d_out;

    // workspace layout (floats)
    float* ws    = (float*)d_ws;
    float* dinv  = ws;                                    // N (deg -> dinv in place)
    float* XW    = dinv + 50048;                          // N*HID
    float* B0    = XW   + (size_t)N_NODES * HID;          // N*HID
    float* B1    = B0   + (size_t)N_NODES * HID;          // N*HID
    float* pool  = B1   + (size_t)N_NODES * HID;          // G*F_OUT
    float* cnt   = pool + N_GRAPHS * F_OUT;               // G (padded)
    float* logit = cnt  + 64;                             // G*OUT_DIM

    // degrees -> dinv
    fill_f32<<<cdiv(N_NODES, 256), 256, 0, stream>>>(dinv, 0.0f, N_NODES);
    deg_kernel<<<cdiv(N_EDGES, 256), 256, 0, stream>>>(dst, dinv, N_EDGES);
    dinv_kernel<<<cdiv(N_NODES, 256), 256, 0, stream>>>(dinv, N_NODES);

    // 4 GCN layers (ping-pong agg buffers; XW reused)
    run_layer(stream, x,  W0, b0, F_IN, HID,   XW, B0, src, dst, dinv);
    run_layer(stream, B0, W1, b1, HID,  HID,   XW, B1, src, dst, dinv);
    run_layer(stream, B1, W2, b2, HID,  HID,   XW, B0, src, dst, dinv);
    run_layer(stream, B0, W3, b3, HID,  F_OUT, XW, B1, src, dst, dinv);

    // global mean pool
    fill_f32<<<cdiv(N_GRAPHS * F_OUT + 64, 256), 256, 0, stream>>>(pool, 0.0f, N_GRAPHS * F_OUT + 64);
    count_kernel<<<cdiv(N_NODES, 256), 256, 0, stream>>>(batch, cnt);
    int poolN = N_NODES * (F_OUT >> 2);
    pool_kernel<<<cdiv(poolN, 256), 256, 0, stream>>>(B1, batch, pool);

    // FC head + log_softmax
    head_kernel<<<cdiv(N_GRAPHS * OUT_DIM, 256), 256, 0, stream>>>(pool, cnt, fcw, fcb, logit);
    logsoftmax_kernel<<<1, 64, 0, stream>>>(logit, out);
}